// HeteroProjectionGNN_85495618994897
// MI455X (gfx1250) — compile-verified
//
#include <hip/hip_runtime.h>

typedef __attribute__((ext_vector_type(2))) float v2f;
typedef __attribute__((ext_vector_type(8))) float v8f;

#define N_PERSON_C 140000
#define N_DISEASE_C 10000
#define N_NODES_C (N_PERSON_C + N_DISEASE_C)   // 150000 = 16 * 9375
#define N_EDGES_C 1200000
#define D_PERSON_C 128
#define D_DISEASE_C 32
#define H_DIM 64

// ---------------------------------------------------------------------------
// Async-stage W[k*64+n] (row-major KxN) into LDS *transposed* as sWt[n*K+k].
// Uses CDNA5 GLOBAL_LOAD_ASYNC_TO_LDS_B32: per-lane global address, per-lane
// LDS byte offset -> the async engine performs the transpose for free.
// The __shared__ array is the only LDS object in each kernel, so its base
// offset is 0 and LDS addresses are plain integers.
// ---------------------------------------------------------------------------
template <int K>
__device__ __forceinline__ void stage_weights_async_T(const float* __restrict__ W)
{
  for (int i = threadIdx.x; i < K * H_DIM; i += 128) {
    const int k = i >> 6;               // source row    (K dim)
    const int n = i & 63;               // source column (N dim)
    const unsigned lds_off = (unsigned)(n * K + k) * 4u;
    const unsigned long long gaddr = (unsigned long long)(const void*)(W + i);
    asm volatile("global_load_async_to_lds_b32 %0, %1, off"
                 :: "v"(lds_off), "v"(gaddr) : "memory");
  }
  asm volatile("s_wait_asynccnt 0x0" ::: "memory");
  __syncthreads();
}

// ---------------------------------------------------------------------------
// One wave computes a full 16x64 output strip: 4 independent f32 WMMA
// accumulators (one per 16-col tile) -> 4-way ILP per K-step and 4x reuse of
// the A fragment. Block = 128 threads = 4 waves; grid-stride over row tiles
// so the LDS weight stage is amortized over many tiles.
// A-fragment layout (ISA 16x4 f32): lane&15 = M row, (lane>>4)*2 = K pair.
// ---------------------------------------------------------------------------
template <int K, bool RELU>
__device__ __forceinline__ void gemm_strip_loop(
    const float* __restrict__ A, int ldA,          // [rows, ldA]
    const float* __restrict__ sWt,                 // LDS, transposed [n*K+k]
    const float* __restrict__ bias,                // [64]
    float* __restrict__ out,                       // [rows, 64]
    int nTiles)
{
  const int lane = threadIdx.x & 31;
  const int wave = threadIdx.x >> 5;
  const int l15  = lane & 15;
  const int kb   = (lane >> 4) << 1;               // 0 or 2: K-pair select

  // B-column bases and bias are tile-invariant: hoist out of the tile loop.
  const float* bcol0 = &sWt[(0 * 16 + l15) * K];
  const float* bcol1 = &sWt[(1 * 16 + l15) * K];
  const float* bcol2 = &sWt[(2 * 16 + l15) * K];
  const float* bcol3 = &sWt[(3 * 16 + l15) * K];
  float bv[4];
#pragma unroll
  for (int t = 0; t < 4; ++t) bv[t] = bias[t * 16 + l15];

  for (int tile = blockIdx.x * 4 + wave; tile < nTiles; tile += gridDim.x * 4) {
    const float* arow = A + ((long)tile * 16 + l15) * ldA;
    v8f c0 = {0.f,0.f,0.f,0.f,0.f,0.f,0.f,0.f};
    v8f c1 = c0, c2 = c0, c3 = c0;
#pragma unroll
    for (int kk = 0; kk < K; kk += 4) {
      const v2f a  = *reinterpret_cast<const v2f*>(arow  + kk + kb);
      const v2f b0 = *reinterpret_cast<const v2f*>(bcol0 + kk + kb);
      const v2f b1 = *reinterpret_cast<const v2f*>(bcol1 + kk + kb);
      const v2f b2 = *reinterpret_cast<const v2f*>(bcol2 + kk + kb);
      const v2f b3 = *reinterpret_cast<const v2f*>(bcol3 + kk + kb);
      c0 = __builtin_amdgcn_wmma_f32_16x16x4_f32(false, a, false, b0, (short)0, c0, false, false);
      c1 = __builtin_amdgcn_wmma_f32_16x16x4_f32(false, a, false, b1, (short)0, c1, false, false);
      c2 = __builtin_amdgcn_wmma_f32_16x16x4_f32(false, a, false, b2, (short)0, c2, false, false);
      c3 = __builtin_amdgcn_wmma_f32_16x16x4_f32(false, a, false, b3, (short)0, c3, false, false);
    }
    // D layout: VGPR i -> M = i + 8*(lane>=16), N = lane&15 (+16*t).
    const int mBase = tile * 16 + ((lane >> 4) << 3);
    v8f cs[4] = {c0, c1, c2, c3};
#pragma unroll
    for (int t = 0; t < 4; ++t) {
      const int n = t * 16 + l15;
#pragma unroll
      for (int i = 0; i < 8; ++i) {
        float v = cs[t][i] + bv[t];
        if (RELU) v = fmaxf(v, 0.f);
        out[(long)(mBase + i) * H_DIM + n] = v;
      }
    }
  }
}

// Projection kernel (K = 128 persons, K = 32 diseases; pointers pre-offset
// to the segment start; feature row stride stays 128).
template <int K>
__global__ __launch_bounds__(128) void proj_wmma_kernel(
    const float* __restrict__ features,
    const float* __restrict__ W, const float* __restrict__ bias,
    float* __restrict__ h, int nTiles)
{
  __shared__ float sWt[K * H_DIM];
  stage_weights_async_T<K>(W);
  gemm_strip_loop<K, false>(features, D_PERSON_C, sWt, bias, h, nTiles);
}

// GCN dense step: out = agg @ W[64x64] + b, optional ReLU.
template <bool RELU>
__global__ __launch_bounds__(128) void gcn_gemm_wmma_kernel(
    const float* __restrict__ A,
    const float* __restrict__ W, const float* __restrict__ bias,
    float* __restrict__ out, int nTiles)
{
  __shared__ float sWt[H_DIM * H_DIM];
  stage_weights_async_T<H_DIM>(W);
  gemm_strip_loop<H_DIM, RELU>(A, H_DIM, sWt, bias, out, nTiles);
}

// ---------------------------------------------------------------------------
// Helpers: zero fill, degree accumulation, edge norms, edge scatter-add.
// ---------------------------------------------------------------------------
__global__ void zero_kernel(float* __restrict__ p, int n) {
  for (int i = blockIdx.x * blockDim.x + threadIdx.x; i < n;
       i += gridDim.x * blockDim.x)
    p[i] = 0.f;
}

__global__ void degree_kernel(const int* __restrict__ src,
                              const int* __restrict__ dst,
                              float* __restrict__ deg_out,
                              float* __restrict__ deg_in) {
  const int e = blockIdx.x * blockDim.x + threadIdx.x;
  if (e < N_EDGES_C) {
    unsafeAtomicAdd(&deg_out[src[e]], 1.0f);   // global_atomic_add_f32
    unsafeAtomicAdd(&deg_in[dst[e]], 1.0f);
  }
}

__global__ void norm_kernel(const int* __restrict__ src,
                            const int* __restrict__ dst,
                            const float* __restrict__ deg_out,
                            const float* __restrict__ deg_in,
                            float* __restrict__ nrm) {
  const int e = blockIdx.x * blockDim.x + threadIdx.x;
  if (e < N_EDGES_C) {
    const float dso = fmaxf(deg_out[src[e]], 1.0f);
    const float dsi = fmaxf(deg_in[dst[e]], 1.0f);
    nrm[e] = rsqrtf(dso * dsi);   // v_rsq_f32, matches jax.lax.rsqrt
  }
}

// One wave per edge; lane handles columns {lane, lane+32}: two coalesced
// 4B gathers + two hardware f32 atomics per lane.
__global__ __launch_bounds__(256) void scatter_kernel(
    const float* __restrict__ h, const int* __restrict__ src,
    const int* __restrict__ dst, const float* __restrict__ nrm,
    float* __restrict__ agg)
{
  const int lane = threadIdx.x & 31;
  const int wid  = (blockIdx.x * blockDim.x + threadIdx.x) >> 5;
  const int nw   = (gridDim.x * blockDim.x) >> 5;
  for (int e = wid; e < N_EDGES_C; e += nw) {
    const int   s = src[e];
    const int   d = dst[e];
    const float w = nrm[e];
    const float* hs = h + (long)s * H_DIM;
    float*       ad = agg + (long)d * H_DIM;
    unsafeAtomicAdd(&ad[lane],      hs[lane]      * w);
    unsafeAtomicAdd(&ad[lane + 32], hs[lane + 32] * w);
  }
}

// ---------------------------------------------------------------------------
extern "C" void kernel_launch(void* const* d_in, const int* in_sizes, int n_in,
                              void* d_out, int out_size, void* d_ws, size_t ws_size,
                              hipStream_t stream)
{
  const float* features = (const float*)d_in[0];
  const int*   src      = (const int*)d_in[1];
  const int*   dst      = (const int*)d_in[2];
  const float* Wp       = (const float*)d_in[3];
  const float* bp       = (const float*)d_in[4];
  const float* Wd       = (const float*)d_in[5];
  const float* bd       = (const float*)d_in[6];
  const float* gcn_W    = (const float*)d_in[7];   // [2,64,64]
  const float* gcn_b    = (const float*)d_in[8];   // [2,64]

  float* h = (float*)d_out;                        // [N, 64] ping-pongs here

  // Workspace layout (floats): agg[N*64] | deg_in[N] | deg_out[N] | norm[E]
  float* agg     = (float*)d_ws;
  float* deg_in  = agg + (size_t)N_NODES_C * H_DIM;
  float* deg_out = deg_in + N_NODES_C;
  float* nrm     = deg_out + N_NODES_C;

  // 1) Hetero projection -> h (WMMA f32; per-type segment, both 16-aligned)
  proj_wmma_kernel<D_PERSON_C><<<512, 128, 0, stream>>>(
      features, Wp, bp, h, N_PERSON_C / 16);
  proj_wmma_kernel<D_DISEASE_C><<<160, 128, 0, stream>>>(
      features + (size_t)N_PERSON_C * D_PERSON_C, Wd, bd,
      h + (size_t)N_PERSON_C * H_DIM, N_DISEASE_C / 16);

  // 2) Degrees and per-edge symmetric norm
  zero_kernel<<<1024, 256, 0, stream>>>(deg_in, 2 * N_NODES_C);
  degree_kernel<<<(N_EDGES_C + 255) / 256, 256, 0, stream>>>(src, dst, deg_out, deg_in);
  norm_kernel<<<(N_EDGES_C + 255) / 256, 256, 0, stream>>>(src, dst, deg_out, deg_in, nrm);

  // 3) Two GCN layers: scatter-add then dense WMMA step
  for (int l = 0; l < 2; ++l) {
    zero_kernel<<<4096, 256, 0, stream>>>(agg, N_NODES_C * H_DIM);
    scatter_kernel<<<(N_EDGES_C + 7) / 8, 256, 0, stream>>>(h, src, dst, nrm, agg);
    if (l == 0)
      gcn_gemm_wmma_kernel<true><<<1024, 128, 0, stream>>>(
          agg, gcn_W, gcn_b, h, N_NODES_C / 16);
    else
      gcn_gemm_wmma_kernel<false><<<1024, 128, 0, stream>>>(
          agg, gcn_W + H_DIM * H_DIM, gcn_b + H_DIM, h, N_NODES_C / 16);
  }
}